// GCNLinkPredictor_21131239096607
// MI455X (gfx1250) — compile-verified
//
#include <hip/hip_runtime.h>
#include <hip/hip_bf16.h>

// ---------------------------------------------------------------------------
// GCN link predictor for MI455X (gfx1250, wave32).
//  - fp32 WMMA (V_WMMA_F32_16X16X4_F32) for the two 50000x128 @ 128x128 GEMMs,
//    with W staged transposed into LDS (bank-conflict-free, ds_load_b64 frags)
//  - native global_atomic_add_f32 for edge scatter (L2-resident, 192MB L2)
//  - wave32 shuffle reduction for the 256-wide link-decode dot products
// ---------------------------------------------------------------------------

typedef __attribute__((ext_vector_type(2))) float v2f;
typedef __attribute__((ext_vector_type(8))) float v8f;

#define DH 128        // hidden / input feature dim
#define WT_STRIDE 132 // padded LDS row stride (132 % 64 = 4 -> conflict-free)

__device__ __forceinline__ void atomAddF(float* p, float v) {
  __hip_atomic_fetch_add(p, v, __ATOMIC_RELAXED, __HIP_MEMORY_SCOPE_AGENT);
}

// ---- degree / normalization ------------------------------------------------

__global__ void k_deg_init(float* __restrict__ deg, int n) {
  int i = blockIdx.x * blockDim.x + threadIdx.x;
  if (i < n) deg[i] = 1.0f;  // self-loop weight
}

__global__ void k_deg_scatter(float* __restrict__ deg,
                              const int* __restrict__ ei,   // [2, E] flat
                              const float* __restrict__ ew, int E) {
  int e = blockIdx.x * blockDim.x + threadIdx.x;
  if (e < E) atomAddF(&deg[ei[E + e]], ew[e]);  // dst row
}

__global__ void k_dinv(float* __restrict__ deg, int n) {
  int i = blockIdx.x * blockDim.x + threadIdx.x;
  if (i < n) {
    float d = deg[i];
    deg[i] = (d > 0.0f) ? rsqrtf(d) : 0.0f;  // in place: deg -> dinv
  }
}

// ---- GEMM: C[M,128] = A[M,128] @ W[128,128] via V_WMMA_F32_16X16X4_F32 -----
// One wave computes a 16x128 output stripe (8 tiles of 16x16).
// W is transposed into LDS once per block: lds_wt[n*132 + k] = W[k*128 + n],
// so each B fragment (k0, k0+1 for fixed n) is one 8-byte ds_load.
// Layouts per CDNA5 ISA 7.12.2 (32-bit A 16x4, 32-bit C/D 16x16).

__global__ void __launch_bounds__(256)
k_gemm_wmma(const float* __restrict__ A, const float* __restrict__ W,
            float* __restrict__ C, int M) {
  __shared__ float lds_wt[DH * WT_STRIDE];  // 67.6 KB of the WGP's 320 KB

  // cooperative transposed load of W (coalesced global reads)
  for (int idx = threadIdx.x; idx < DH * DH; idx += blockDim.x) {
    int k = idx >> 7;
    int n = idx & 127;
    lds_wt[n * WT_STRIDE + k] = W[idx];
  }
  __syncthreads();

  const int lane = threadIdx.x & 31;
  const int wv   = threadIdx.x >> 5;
  const int mt   = blockIdx.x * (blockDim.x >> 5) + wv;  // wave-uniform
  if (mt * 16 >= M) return;  // after barrier; EXEC stays all-1s for WMMA

  const int l16  = lane & 15;
  const int half = lane >> 4;
  const long rowA = (long)(mt * 16 + l16);

  v8f acc[8] = {};  // 8 N-tiles of the 16x128 stripe

#pragma unroll 4
  for (int kt = 0; kt < 32; ++kt) {
    const int k0 = kt * 4 + half * 2;                 // A/B frag K index
    const v2f a = *(const v2f*)(A + rowA * DH + k0);  // global_load_b64
#pragma unroll
    for (int nt = 0; nt < 8; ++nt) {
      const v2f b =
          *(const v2f*)(lds_wt + (nt * 16 + l16) * WT_STRIDE + k0);  // ds b64
      acc[nt] = __builtin_amdgcn_wmma_f32_16x16x4_f32(
          /*neg_a=*/false, a, /*neg_b=*/false, b,
          /*c_mod=*/(short)0, acc[nt], /*reuse_a=*/false, /*reuse_b=*/false);
    }
  }

#pragma unroll
  for (int nt = 0; nt < 8; ++nt) {
#pragma unroll
    for (int i = 0; i < 8; ++i) {
      // C/D layout: VGPR i -> row i (lanes 0-15) / row i+8 (lanes 16-31)
      C[(long)(mt * 16 + half * 8 + i) * DH + nt * 16 + l16] = acc[nt][i];
    }
  }
}

// ---- self-loop init: out[i,:] = h[i,:] * dinv[i]^2 -------------------------

__global__ void k_selfloop(const float* __restrict__ h,
                           const float* __restrict__ dinv,
                           float* __restrict__ out, int n) {
  int t = blockIdx.x * blockDim.x + threadIdx.x;
  if (t >= n * 32) return;
  int node = t >> 5;
  int c    = (t & 31) * 4;
  float di = dinv[node];
  float sc = di * di;
  float4 v = *(const float4*)(h + (long)node * DH + c);
  v.x *= sc; v.y *= sc; v.z *= sc; v.w *= sc;
  *(float4*)(out + (long)node * DH + c) = v;
}

// ---- edge scatter: out[dst,:] += h[src,:] * (dinv[s]*w*dinv[d]) ------------
// One wave per edge; lane owns a float4 feature chunk -> 4 f32 atomics.

__global__ void k_edge_scatter(const float* __restrict__ h,
                               const int* __restrict__ ei,  // [2, E] flat
                               const float* __restrict__ ew,
                               const float* __restrict__ dinv,
                               float* __restrict__ out, int E) {
  const int lane = threadIdx.x & 31;
  const int wv   = threadIdx.x >> 5;
  const int e    = blockIdx.x * (blockDim.x >> 5) + wv;
  if (e >= E) return;
  const int s = ei[e];
  const int d = ei[E + e];
  const float norm = dinv[s] * ew[e] * dinv[d];
  const int c = lane * 4;
  float4 v = *(const float4*)(h + (long)s * DH + c);
  float* o = out + (long)d * DH + c;
  atomAddF(o + 0, v.x * norm);
  atomAddF(o + 1, v.y * norm);
  atomAddF(o + 2, v.z * norm);
  atomAddF(o + 3, v.w * norm);
}

// ---- bias (+ optional ReLU), in place --------------------------------------

__global__ void k_bias_act(float* __restrict__ z, const float* __restrict__ b,
                           int n, int do_relu) {
  int t = blockIdx.x * blockDim.x + threadIdx.x;
  if (t >= n * 32) return;
  int node = t >> 5;
  int c    = (t & 31) * 4;
  float4 v  = *(const float4*)(z + (long)node * DH + c);
  float4 bb = *(const float4*)(b + c);
  v.x += bb.x; v.y += bb.y; v.z += bb.z; v.w += bb.w;
  if (do_relu) {
    v.x = fmaxf(v.x, 0.0f); v.y = fmaxf(v.y, 0.0f);
    v.z = fmaxf(v.z, 0.0f); v.w = fmaxf(v.w, 0.0f);
  }
  *(float4*)(z + (long)node * DH + c) = v;
}

// ---- decode: score = [z[s], z[d]] . W_link + b -----------------------------
// One wave per eval edge; lane covers 4 features of each endpoint.

__global__ void k_decode(const float* __restrict__ z,
                         const int* __restrict__ ei,  // [2, Eev] flat
                         const float* __restrict__ Wl,
                         const float* __restrict__ bl,
                         float* __restrict__ out, int Eev) {
  const int lane = threadIdx.x & 31;
  const int wv   = threadIdx.x >> 5;
  const int e    = blockIdx.x * (blockDim.x >> 5) + wv;
  if (e >= Eev) return;
  const int s = ei[e];
  const int d = ei[Eev + e];
  const int c = lane * 4;
  float4 zs = *(const float4*)(z + (long)s * DH + c);
  float4 zd = *(const float4*)(z + (long)d * DH + c);
  float4 ws = *(const float4*)(Wl + c);
  float4 wd = *(const float4*)(Wl + DH + c);
  float p = zs.x * ws.x + zs.y * ws.y + zs.z * ws.z + zs.w * ws.w +
            zd.x * wd.x + zd.y * wd.y + zd.z * wd.z + zd.w * wd.w;
#pragma unroll
  for (int off = 16; off > 0; off >>= 1) p += __shfl_xor(p, off, 32);
  if (lane == 0) out[e] = p + bl[0];
}

// ---------------------------------------------------------------------------

extern "C" void kernel_launch(void* const* d_in, const int* in_sizes, int n_in,
                              void* d_out, int out_size, void* d_ws, size_t ws_size,
                              hipStream_t stream) {
  const float* x   = (const float*)d_in[0];
  const int*   ei  = (const int*)d_in[1];
  const float* ew  = (const float*)d_in[2];
  const int*   pei = (const int*)d_in[3];
  const int*   nei = (const int*)d_in[4];
  const float* W1  = (const float*)d_in[5];
  const float* b1  = (const float*)d_in[6];
  const float* W2  = (const float*)d_in[7];
  const float* b2  = (const float*)d_in[8];
  const float* Wl  = (const float*)d_in[9];
  const float* bl  = (const float*)d_in[10];
  float* out = (float*)d_out;

  const int N   = in_sizes[0] / DH;   // 50000
  const int E   = in_sizes[2];        // 640000
  const int Eev = in_sizes[3] / 2;    // 100000

  // workspace: [dinv: N (rounded)] [buf0: N*128] [buf1: N*128]
  float* dinv = (float*)d_ws;
  float* buf0 = dinv + ((N + 63) & ~63);
  float* buf1 = buf0 + (long)N * DH;

  const int T = 256;
  const int gN   = (N + T - 1) / T;
  const int gE   = (E + T - 1) / T;
  const int gNF  = (N * 32 + T - 1) / T;           // float4-chunk grid
  const int gEw  = (E + (T >> 5) - 1) / (T >> 5);  // wave-per-edge grid
  const int gEv  = (Eev + (T >> 5) - 1) / (T >> 5);
  const int mtiles = N / 16;                       // 3125 (N divisible by 16)
  const int gG   = (mtiles + (T >> 5) - 1) / (T >> 5);

  // normalization
  k_deg_init<<<gN, T, 0, stream>>>(dinv, N);
  k_deg_scatter<<<gE, T, 0, stream>>>(dinv, ei, ew, E);
  k_dinv<<<gN, T, 0, stream>>>(dinv, N);

  // layer 1: h1 = x @ W1 ; aggregate ; z1 = relu(agg + b1)
  k_gemm_wmma<<<gG, T, 0, stream>>>(x, W1, buf0, N);
  k_selfloop<<<gNF, T, 0, stream>>>(buf0, dinv, buf1, N);
  k_edge_scatter<<<gEw, T, 0, stream>>>(buf0, ei, ew, dinv, buf1, E);
  k_bias_act<<<gNF, T, 0, stream>>>(buf1, b1, N, 1);

  // layer 2: h2 = z1 @ W2 ; aggregate ; z = agg + b2
  k_gemm_wmma<<<gG, T, 0, stream>>>(buf1, W2, buf0, N);
  k_selfloop<<<gNF, T, 0, stream>>>(buf0, dinv, buf1, N);
  k_edge_scatter<<<gEw, T, 0, stream>>>(buf0, ei, ew, dinv, buf1, E);
  k_bias_act<<<gNF, T, 0, stream>>>(buf1, b2, N, 0);

  // decode pos / neg
  k_decode<<<gEv, T, 0, stream>>>(buf1, pei, Wl, bl, out, Eev);
  k_decode<<<gEv, T, 0, stream>>>(buf1, nei, Wl, bl, out + Eev, Eev);
}